// BasicBlock_83640193122772
// MI455X (gfx1250) — compile-verified
//
#include <hip/hip_runtime.h>
#include <cstddef>

typedef __attribute__((ext_vector_type(2))) float v2f;
typedef __attribute__((ext_vector_type(8))) float v8f;

#define NSEG 4096
#define EPS_BN 1e-5f

#define F_RELU     1
#define F_MUL      2
#define F_RES_PRE  4
#define F_RES_POST 8

#if __has_builtin(__builtin_amdgcn_global_load_async_to_lds_b128) && \
    __has_builtin(__builtin_amdgcn_s_wait_asynccnt)
#define HAVE_ASYNC_LDS 1
#define AS1 __attribute__((address_space(1)))
#define AS3 __attribute__((address_space(3)))
typedef __attribute__((__vector_size__(4 * sizeof(int)))) int v4i_t;
#endif

// ---------------------------------------------------------------------------
// GEMM: Y[N,128] = A[N,K] @ Bw[K,128], K in {128,256} (template). A split as
// A0|A1 when K==256. Optional fused per-channel sum/sumsq for BatchNorm.
// Block: 256 threads = 8 waves; each wave computes a 16x128 strip with
// V_WMMA_F32_16X16X4_F32 (fp32 in, fp32 accumulate -> reference numerics).
// B staged in LDS pair-interleaved: element (k,c) -> sB[(k>>1)*288 + c*2 +
// (k&1)], so each B fragment is one aligned ds_load_b64; pair-row stride 288
// floats puts lanes 0-15 / 16-31 on disjoint bank halves.
// ---------------------------------------------------------------------------
template <int K>
__global__ __launch_bounds__(256) void k_gemm_wmma(
    const float* __restrict__ A0, const float* __restrict__ A1,
    const float* __restrict__ Bw,
    float* __restrict__ Y, float* __restrict__ stats, int N)
{
    __shared__ float sB[32 * 288];   // one 64-K chunk, pair-interleaved
    __shared__ float sSum[128];
    __shared__ float sSq[128];

    const int tid  = threadIdx.x;
    const int lane = tid & 31;
    const int wave = tid >> 5;
    const int m    = lane & 15;      // row-in-tile (A) / col-in-tile (B,C)
    const int kh   = lane >> 4;      // K-half select (A/B: K = kh*2 + {0,1})
    const int rowBase = blockIdx.x * 128 + wave * 16;
    const int row  = rowBase + m;
    // clamp instead of branch: OOB rows compute dead values, never stored,
    // excluded from stats -> no exec divergence in the inner loop
    const int rowC = (row < N) ? row : (N - 1);
    const float* __restrict__ Arow0 = A0 + (size_t)rowC * 128;
    const float* __restrict__ Arow1 = A1 + (size_t)rowC * 128;

    v8f acc[8];
#pragma unroll
    for (int t = 0; t < 8; ++t)
#pragma unroll
        for (int r = 0; r < 8; ++r) acc[t][r] = 0.0f;

#pragma unroll
    for (int kc = 0; kc < K; kc += 64) {
        // cooperative B chunk stage: 64x128 floats, pair-interleaved scatter
        for (int i = tid; i < 64 * 32; i += 256) {
            const int r  = i >> 5;               // k within chunk
            const int c4 = (i & 31) << 2;        // col group
            const float4 v = *(const float4*)(Bw + (size_t)(kc + r) * 128 + c4);
            float* d = &sB[(r >> 1) * 288 + c4 * 2 + (r & 1)];
            d[0] = v.x; d[2] = v.y; d[4] = v.z; d[6] = v.w;
        }
        __syncthreads();
#pragma unroll 4
        for (int kt = 0; kt < 16; ++kt) {
            const int kg = kc + kt * 4 + kh * 2;  // even; folds at compile time
            const float2 av = (kg < 128) ? *(const float2*)(Arow0 + kg)
                                         : *(const float2*)(Arow1 + (kg - 128));
            v2f a; a[0] = av.x; a[1] = av.y;
            const int prow = (kt * 2 + kh) * 288;
#pragma unroll
            for (int ct = 0; ct < 8; ++ct) {
                const v2f b = *(const v2f*)(&sB[prow + (ct * 16 + m) * 2]);
                acc[ct] = __builtin_amdgcn_wmma_f32_16x16x4_f32(
                    false, a, false, b, (short)0, acc[ct], false, false);
            }
        }
        __syncthreads();
    }

    // C/D layout: VGPR r -> lanes0-15: M=r, lanes16-31: M=r+8; N = m
    const int rOff = kh * 8;
#pragma unroll
    for (int ct = 0; ct < 8; ++ct) {
        const int col = ct * 16 + m;
#pragma unroll
        for (int r = 0; r < 8; ++r) {
            const int rr = rowBase + rOff + r;
            if (rr < N) Y[(size_t)rr * 128 + col] = acc[ct][r];
        }
    }

    if (stats) {   // uniform pointer -> uniform barriers
        if (tid < 128) { sSum[tid] = 0.0f; sSq[tid] = 0.0f; }
        __syncthreads();
#pragma unroll
        for (int ct = 0; ct < 8; ++ct) {
            float s = 0.0f, q = 0.0f;
#pragma unroll
            for (int r = 0; r < 8; ++r) {
                const int rr = rowBase + rOff + r;
                if (rr < N) { const float v = acc[ct][r]; s += v; q += v * v; }
            }
            atomicAdd(&sSum[ct * 16 + m], s);   // ds_add_f32
            atomicAdd(&sSq[ct * 16 + m], q);
        }
        __syncthreads();
        if (tid < 128) {
            atomicAdd(&stats[tid], sSum[tid]);       // one global atomic/ch/block
            atomicAdd(&stats[128 + tid], sSq[tid]);
        }
    }
}

// ---------------------------------------------------------------------------
// Elementwise BN epilogue with fused relu / multiplier / residual variants.
// ---------------------------------------------------------------------------
__global__ __launch_bounds__(256) void k_bnmix(
    const float* __restrict__ in, float* __restrict__ out,
    const float* __restrict__ stats,
    const float* __restrict__ g, const float* __restrict__ b,
    const float* __restrict__ mul, const float* __restrict__ res,
    int flags, int N)
{
    const size_t t = (size_t)blockIdx.x * 256 + threadIdx.x;
    if (t >= (size_t)N * 128) return;
    const int c = (int)(t & 127);
    const float invN = 1.0f / (float)N;
    const float mean = stats[c] * invN;
    const float var  = stats[128 + c] * invN - mean * mean;
    const float rs   = rsqrtf(var + EPS_BN);
    float v = (in[t] - mean) * rs * g[c] + b[c];
    if (flags & F_RES_PRE)  v += res[t];
    if (flags & F_RELU)     v = fmaxf(v, 0.0f);
    if (flags & F_MUL)      v *= mul[t];
    if (flags & F_RES_POST) v += res[t];
    out[t] = v;
}

__global__ __launch_bounds__(256) void k_count(const int* __restrict__ cl,
                                               float* __restrict__ cnt, int N)
{
    const int n = blockIdx.x * 256 + threadIdx.x;
    if (n < N) atomicAdd(&cnt[cl[n]], 1.0f);
}

// seg[cl[n], c] += X[n, c]; one thread handles 4 channels (float4 load)
__global__ __launch_bounds__(256) void k_segsum(const float* __restrict__ X,
                                                const int* __restrict__ cl,
                                                float* __restrict__ seg, int N)
{
    const size_t t = (size_t)blockIdx.x * 256 + threadIdx.x;
    if (t >= (size_t)N * 32) return;
    const int n = (int)(t >> 5);
    const int q = (int)(t & 31);
    const float4 v = *(const float4*)(X + (size_t)n * 128 + q * 4);
    float* d = seg + (size_t)cl[n] * 128 + q * 4;
    atomicAdd(d + 0, v.x); atomicAdd(d + 1, v.y);
    atomicAdd(d + 2, v.z); atomicAdd(d + 3, v.w);
}

__global__ __launch_bounds__(256) void k_center(float* __restrict__ Y,
                                                const float* __restrict__ seg,
                                                const float* __restrict__ cnt,
                                                const int* __restrict__ cl, int N)
{
    const size_t t = (size_t)blockIdx.x * 256 + threadIdx.x;
    if (t >= (size_t)N * 128) return;
    const int n = (int)(t >> 7), c = (int)(t & 127);
    const int s = cl[n];
    Y[t] -= seg[(size_t)s * 128 + c] / fmaxf(cnt[s], 1.0f);
}

__global__ __launch_bounds__(256) void k_maxred(const float* __restrict__ X,
                                                float* __restrict__ part, size_t count)
{
    __shared__ float s[256];
    float m = -3.402823466e38f;
    for (size_t i = (size_t)blockIdx.x * 256 + threadIdx.x; i < count;
         i += (size_t)gridDim.x * 256)
        m = fmaxf(m, X[i]);
    s[threadIdx.x] = m; __syncthreads();
    for (int o = 128; o > 0; o >>= 1) {
        if ((int)threadIdx.x < o) s[threadIdx.x] = fmaxf(s[threadIdx.x], s[threadIdx.x + o]);
        __syncthreads();
    }
    if (threadIdx.x == 0) part[blockIdx.x] = s[0];
}

__global__ __launch_bounds__(256) void k_maxfinal(const float* __restrict__ part,
                                                  int n, float* __restrict__ gmax)
{
    __shared__ float s[256];
    float m = -3.402823466e38f;
    for (int i = threadIdx.x; i < n; i += 256) m = fmaxf(m, part[i]);
    s[threadIdx.x] = m; __syncthreads();
    for (int o = 128; o > 0; o >>= 1) {
        if ((int)threadIdx.x < o) s[threadIdx.x] = fmaxf(s[threadIdx.x], s[threadIdx.x + o]);
        __syncthreads();
    }
    if (threadIdx.x == 0) gmax[0] = s[0];
}

__global__ __launch_bounds__(256) void k_exp(float* __restrict__ Y,
                                             const float* __restrict__ gmax, int N)
{
    const size_t t = (size_t)blockIdx.x * 256 + threadIdx.x;
    if (t < (size_t)N * 128) Y[t] = expf(Y[t] - gmax[0]);
}

__global__ __launch_bounds__(256) void k_divseg(float* __restrict__ Y,
                                                const float* __restrict__ seg,
                                                const int* __restrict__ cl, int N)
{
    const size_t t = (size_t)blockIdx.x * 256 + threadIdx.x;
    if (t >= (size_t)N * 128) return;
    const int n = (int)(t >> 7), c = (int)(t & 127);
    Y[t] /= (seg[(size_t)cl[n] * 128 + c] + 1e-6f);
}

__global__ __launch_bounds__(256) void k_adp(const float* __restrict__ feat,
                                             const float* __restrict__ adpW,
                                             float* __restrict__ adp, int N)
{
    const int n = blockIdx.x * 256 + threadIdx.x;
    if (n >= N) return;
    const float* f = feat + (size_t)n * 128;
    float a0 = 0.f, a1 = 0.f, a2 = 0.f;
    for (int c = 0; c < 128; ++c) {
        const float v = f[c];
        a0 += v * adpW[c * 3 + 0];
        a1 += v * adpW[c * 3 + 1];
        a2 += v * adpW[c * 3 + 2];
    }
    const float mx = fmaxf(a0, fmaxf(a1, a2));
    const float e0 = expf(a0 - mx), e1 = expf(a1 - mx), e2 = expf(a2 - mx);
    const float inv = 1.0f / (e0 + e1 + e2);
    adp[(size_t)n * 3 + 0] = e0 * inv;
    adp[(size_t)n * 3 + 1] = e1 * inv;
    adp[(size_t)n * 3 + 2] = e2 * inv;
}

// agg[n,c] = sum_l adp[n,l] * segF[l][cl_l[n], c]
__global__ __launch_bounds__(256) void k_agg(const float* __restrict__ adp,
                                             const float* __restrict__ segF,
                                             const int* __restrict__ clusters,
                                             float* __restrict__ out, int N)
{
    const size_t t = (size_t)blockIdx.x * 256 + threadIdx.x;
    if (t >= (size_t)N * 128) return;
    const int n = (int)(t >> 7), c = (int)(t & 127);
    float acc = 0.0f;
    for (int l = 0; l < 3; ++l) {
        const int s = clusters[(size_t)l * N + n];
        acc += adp[(size_t)n * 3 + l] *
               segF[(size_t)l * NSEG * 128 + (size_t)s * 128 + c];
    }
    out[t] = acc;
}

// ---------------------------------------------------------------------------
// Grouped submanifold conv: out[n, g*4+o] = sum_k sum_i F[nidx[n,k], g*4+i] *
// W[k,g,i,o]. One wave per point, lane = group; W (54KB) staged in LDS via
// gfx1250 async global->LDS copies when available. Fused BN statistics.
// ---------------------------------------------------------------------------
__global__ __launch_bounds__(256) void k_conv(const float* __restrict__ F,
                                              const int* __restrict__ nidx,
                                              const float* __restrict__ W,
                                              float* __restrict__ out,
                                              float* __restrict__ stats, int N)
{
    __shared__ float sW[27 * 32 * 16];
    __shared__ float sSum[128];
    __shared__ float sSq[128];
    const int tid = threadIdx.x;
#ifdef HAVE_ASYNC_LDS
    // async tensor-style staging: 3456 x b128 copies, VGPR-bypassing
    for (int i = tid; i < (27 * 32 * 16) / 4; i += 256) {
        __builtin_amdgcn_global_load_async_to_lds_b128(
            (AS1 v4i_t*)(W + (size_t)i * 4),
            (AS3 v4i_t*)(sW + (size_t)i * 4),
            0, 0);
    }
    __builtin_amdgcn_s_wait_asynccnt(0);
#else
    for (int i = tid; i < 27 * 32 * 16; i += 256) sW[i] = W[i];
#endif
    if (tid < 128) { sSum[tid] = 0.0f; sSq[tid] = 0.0f; }
    __syncthreads();

    const int wave = tid >> 5, lane = tid & 31;
    const int n = blockIdx.x * 8 + wave;
    if (n < N) {
        const int g = lane;
        float a0 = 0.f, a1 = 0.f, a2 = 0.f, a3 = 0.f;
        for (int k = 0; k < 27; ++k) {
            const int idx = nidx[(size_t)n * 27 + k];
            if (idx >= 0) {
                const float4 f = *(const float4*)(F + (size_t)idx * 128 + g * 4);
                const float* w = &sW[(k * 32 + g) * 16];   // [i][o] 4x4
                a0 += f.x * w[0] + f.y * w[4] + f.z * w[8]  + f.w * w[12];
                a1 += f.x * w[1] + f.y * w[5] + f.z * w[9]  + f.w * w[13];
                a2 += f.x * w[2] + f.y * w[6] + f.z * w[10] + f.w * w[14];
                a3 += f.x * w[3] + f.y * w[7] + f.z * w[11] + f.w * w[15];
            }
        }
        float* o = out + (size_t)n * 128 + g * 4;
        o[0] = a0; o[1] = a1; o[2] = a2; o[3] = a3;
        atomicAdd(&sSum[g * 4 + 0], a0); atomicAdd(&sSq[g * 4 + 0], a0 * a0);
        atomicAdd(&sSum[g * 4 + 1], a1); atomicAdd(&sSq[g * 4 + 1], a1 * a1);
        atomicAdd(&sSum[g * 4 + 2], a2); atomicAdd(&sSq[g * 4 + 2], a2 * a2);
        atomicAdd(&sSum[g * 4 + 3], a3); atomicAdd(&sSq[g * 4 + 3], a3 * a3);
    }
    __syncthreads();
    if (tid < 128) {
        atomicAdd(&stats[tid], sSum[tid]);
        atomicAdd(&stats[128 + tid], sSq[tid]);
    }
}

// ---------------------------------------------------------------------------
extern "C" void kernel_launch(void* const* d_in, const int* in_sizes, int n_in,
                              void* d_out, int out_size, void* d_ws, size_t ws_size,
                              hipStream_t stream)
{
    const float* feat    = (const float*)d_in[0];
    const float* lw_W    = (const float*)d_in[1];
    const float* lw_g    = (const float*)d_in[2];
    const float* lw_b    = (const float*)d_in[3];
    const float* w_W     = (const float*)d_in[4];
    const float* proj_W  = (const float*)d_in[5];
    const float* proj_g  = (const float*)d_in[6];
    const float* proj_b  = (const float*)d_in[7];
    const float* adp_W   = (const float*)d_in[8];
    const float* fuse_W  = (const float*)d_in[9];
    const float* fuse_g  = (const float*)d_in[10];
    const float* fuse_b  = (const float*)d_in[11];
    const float* conv1_W = (const float*)d_in[12];
    const float* n1_g    = (const float*)d_in[13];
    const float* n1_b    = (const float*)d_in[14];
    const float* conv2_W = (const float*)d_in[15];
    const float* n2_g    = (const float*)d_in[16];
    const float* n2_b    = (const float*)d_in[17];
    const int*   clusters= (const int*)d_in[18];
    const int*   nidx    = (const int*)d_in[19];
    float* out = (float*)d_out;

    const int N = in_sizes[0] / 128;
    const size_t NC = (size_t)N * 128;

    // workspace layout (floats)
    float* B0   = (float*)d_ws;
    float* B1   = B0 + NC;
    float* B2   = B1 + NC;
    float* segF = B2 + NC;                          // 3 * NSEG * 128
    float* segA = segF + (size_t)3 * NSEG * 128;    // NSEG * 128
    float* cnt  = segA + (size_t)NSEG * 128;        // NSEG
    float* stats= cnt + NSEG;                       // 256
    float* gmax = stats + 256;                      // 1
    float* part = gmax + 1;                         // 1024
    float* adp  = part + 1024;                      // N * 3

    const int gGemm = (N + 127) / 128;
    const int gElem = (int)((NC + 255) / 256);
    const int gSeg  = (int)(((size_t)N * 32 + 255) / 256);
    const int gPt   = (N + 255) / 256;
    const int gConv = (N + 7) / 8;
    const size_t segBytes = (size_t)NSEG * 128 * sizeof(float);

    for (int i = 0; i < 3; ++i) {
        const int* cl = clusters + (size_t)i * N;
        (void)hipMemsetAsync(cnt, 0, NSEG * sizeof(float), stream);
        k_count<<<gPt, 256, 0, stream>>>(cl, cnt, N);
        // pw = relu(bn(feat @ lw_W[i]))
        (void)hipMemsetAsync(stats, 0, 256 * sizeof(float), stream);
        k_gemm_wmma<128><<<gGemm, 256, 0, stream>>>(feat, feat,
                                                    lw_W + (size_t)i * 128 * 128,
                                                    B0, stats, N);
        k_bnmix<<<gElem, 256, 0, stream>>>(B0, B0, stats, lw_g + i * 128, lw_b + i * 128,
                                           nullptr, nullptr, F_RELU, N);
        // center per cluster
        (void)hipMemsetAsync(segA, 0, segBytes, stream);
        k_segsum<<<gSeg, 256, 0, stream>>>(B0, cl, segA, N);
        k_center<<<gElem, 256, 0, stream>>>(B0, segA, cnt, cl, N);
        // pw @ w_W[i], scatter-softmax
        k_gemm_wmma<128><<<gGemm, 256, 0, stream>>>(B0, B0,
                                                    w_W + (size_t)i * 128 * 128,
                                                    B1, nullptr, N);
        k_maxred<<<1024, 256, 0, stream>>>(B1, part, NC);
        k_maxfinal<<<1, 256, 0, stream>>>(part, 1024, gmax);
        k_exp<<<gElem, 256, 0, stream>>>(B1, gmax, N);
        (void)hipMemsetAsync(segA, 0, segBytes, stream);
        k_segsum<<<gSeg, 256, 0, stream>>>(B1, cl, segA, N);
        k_divseg<<<gElem, 256, 0, stream>>>(B1, segA, cl, N);
        // pfeat = relu(bn(feat @ proj_W[i])) * pw ; feats[i] = segsum(pfeat)
        (void)hipMemsetAsync(stats, 0, 256 * sizeof(float), stream);
        k_gemm_wmma<128><<<gGemm, 256, 0, stream>>>(feat, feat,
                                                    proj_W + (size_t)i * 128 * 128,
                                                    B2, stats, N);
        k_bnmix<<<gElem, 256, 0, stream>>>(B2, B2, stats, proj_g + i * 128, proj_b + i * 128,
                                           B1, nullptr, F_RELU | F_MUL, N);
        (void)hipMemsetAsync(segF + (size_t)i * NSEG * 128, 0, segBytes, stream);
        k_segsum<<<gSeg, 256, 0, stream>>>(B2, cl, segF + (size_t)i * NSEG * 128, N);
    }

    // adp = softmax(feat @ adp_W); f_last; agg; fuse
    k_adp<<<gPt, 256, 0, stream>>>(feat, adp_W, adp, N);
    (void)hipMemsetAsync(stats, 0, 256 * sizeof(float), stream);
    k_gemm_wmma<128><<<gGemm, 256, 0, stream>>>(feat, feat,
                                                proj_W + (size_t)3 * 128 * 128,
                                                B0, stats, N);
    k_bnmix<<<gElem, 256, 0, stream>>>(B0, B0, stats, proj_g + 3 * 128, proj_b + 3 * 128,
                                       nullptr, nullptr, F_RELU, N);
    k_agg<<<gElem, 256, 0, stream>>>(adp, segF, clusters, B1, N);
    (void)hipMemsetAsync(stats, 0, 256 * sizeof(float), stream);
    k_gemm_wmma<256><<<gGemm, 256, 0, stream>>>(B0, B1, fuse_W, B2, stats, N);
    // fused = relu(bn(...)) + feat   (B2 = fused = res)
    k_bnmix<<<gElem, 256, 0, stream>>>(B2, B2, stats, fuse_g, fuse_b,
                                       nullptr, feat, F_RELU | F_RES_POST, N);
    // conv1 -> bn ; conv2 -> relu(bn + fused)
    (void)hipMemsetAsync(stats, 0, 256 * sizeof(float), stream);
    k_conv<<<gConv, 256, 0, stream>>>(B2, nidx, conv1_W, B0, stats, N);
    k_bnmix<<<gElem, 256, 0, stream>>>(B0, B0, stats, n1_g, n1_b, nullptr, nullptr, 0, N);
    (void)hipMemsetAsync(stats, 0, 256 * sizeof(float), stream);
    k_conv<<<gConv, 256, 0, stream>>>(B0, nidx, conv2_W, B1, stats, N);
    k_bnmix<<<gElem, 256, 0, stream>>>(B1, out, stats, n2_g, n2_b,
                                       nullptr, B2, F_RELU | F_RES_PRE, N);

    (void)n_in; (void)out_size; (void)ws_size;
}